// ListMLE_84112639525723
// MI455X (gfx1250) — compile-verified
//
#include <hip/hip_runtime.h>
#include <hip/hip_bf16.h>
#include <math.h>

#define EPS      1e-10f
#define N_ITEMS  1024
#define THREADS  128      // 4 wave32s per workgroup, one workgroup per row
#define WAVES    4
#define CHUNK    256      // elements handled per wave (16 segments x 16)

// pad 1 dword per 16 elements: lane stride becomes 17 dwords -> conflict-free
#define PADDED(p)  ((p) + ((p) >> 4))
#define SS_SIZE    (N_ITEMS + (N_ITEMS >> 4))

typedef float v2f __attribute__((ext_vector_type(2)));
typedef float v8f __attribute__((ext_vector_type(8)));

// One workgroup per row. Suffix-cumsum of exp(scores_sorted) computed as
// D = T(16x16 upper-triangular ones) x V(16x16 values) on the WMMA pipe,
// chained as 4 x V_WMMA_F32_16X16X4_F32 (K=4 slices accumulated through C).
__global__ __launch_bounds__(THREADS)
void ListMLE_row_kernel(const float* __restrict__ scores,
                        const int*   __restrict__ rankings,
                        float* __restrict__ row_loss)
{
    __shared__ __align__(16) float s_scores[N_ITEMS];
    __shared__ __align__(16) float s_ss[SS_SIZE];   // gathered scores, padded
    __shared__ float s_chunk_tot[WAVES];
    __shared__ float s_wave_loss[WAVES];

    const int row = blockIdx.x;
    const int tid = threadIdx.x;
    const float4* srow4 = reinterpret_cast<const float4*>(scores   + (size_t)row * N_ITEMS);
    const int4*   rrow4 = reinterpret_cast<const int4*>  (rankings + (size_t)row * N_ITEMS);

    // 1) stage scores row into LDS: B128 loads, B128 LDS stores (coalesced)
    #pragma unroll
    for (int i = tid; i < N_ITEMS / 4; i += THREADS) {
        *reinterpret_cast<float4*>(&s_scores[4 * i]) = srow4[i];
    }
    __syncthreads();

    // 2) gather by rankings (B128 index loads, LDS-resident gather, padded store)
    #pragma unroll
    for (int i = tid; i < N_ITEMS / 4; i += THREADS) {
        const int4 r = rrow4[i];
        const int p = 4 * i;
        s_ss[PADDED(p + 0)] = s_scores[r.x & (N_ITEMS - 1)];
        s_ss[PADDED(p + 1)] = s_scores[r.y & (N_ITEMS - 1)];
        s_ss[PADDED(p + 2)] = s_scores[r.z & (N_ITEMS - 1)];
        s_ss[PADDED(p + 3)] = s_scores[r.w & (N_ITEMS - 1)];
    }
    __syncthreads();

    const int  wave  = tid >> 5;
    const int  lane  = tid & 31;
    const int  nCol  = lane & 15;           // segment (column) index
    const bool hi    = lane >= 16;
    const int  kBase = hi ? 2 : 0;          // K pair held by this half-wave
    const int  cbase = wave * CHUNK;
    // padded base for this lane's column: PADDED(cbase + 16*nCol + k) = bcol + k  (k<16)
    const int  bcol  = cbase + wave * 16 + 17 * nCol;

    // 3) segmented suffix scan: acc[k][n] = sum_{j>=k} exp(ss[cbase+16n+j])
    v8f acc = {};
    #pragma unroll
    for (int c = 0; c < 4; ++c) {
        const int k0 = 4 * c + kBase;
        v2f a, b;
        // A = T: A[m][k] = (k >= m), m = lane%16
        a.x = (k0     >= nCol) ? 1.0f : 0.0f;
        a.y = (k0 + 1 >= nCol) ? 1.0f : 0.0f;
        // B = V rows k0,k0+1 for column n (conflict-free padded reads)
        b.x = __expf(s_ss[bcol + k0]);
        b.y = __expf(s_ss[bcol + k0 + 1]);
        acc = __builtin_amdgcn_wmma_f32_16x16x4_f32(false, a, false, b,
                                                    (short)0, acc, false, false);
    }

    // segment totals t[n] = D[0][n] live on lanes 0..15 of acc[0]; mirror to hi half
    float t = __shfl(acc[0], nCol, 32);
    // inclusive suffix scan across the 16 segments (mirrored on both halves)
    float incl = t;
    #pragma unroll
    for (int d = 1; d < 16; d <<= 1) {
        float o = __shfl_down(incl, d, 32);
        if (nCol + d < 16) incl += o;
    }
    const float off = incl - t;             // exclusive suffix offset for column n

    // cross-wave (chunk) offsets
    if (lane == 0) s_chunk_tot[wave] = incl;   // incl at n==0 == chunk total
    __syncthreads();
    float cross = 0.0f;
    for (int w = 0; w < WAVES; ++w)
        if (w > wave) cross += s_chunk_tot[w];
    const float addv = off + cross;

    // 4) per-item loss; exclude last position (p == N-1)
    float part = 0.0f;
    #pragma unroll
    for (int r = 0; r < 8; ++r) {
        const int k = r + (hi ? 8 : 0);     // D VGPR r: M=r (lo) / M=r+8 (hi)
        const int p = cbase + 16 * nCol + k;
        const float S  = acc[r] + addv;     // full suffix sum at position p
        const float ss = s_ss[bcol + k];    // same element, padded address
        if (p != N_ITEMS - 1) part += __logf(S + EPS) - ss;
    }

    // wave reduce, then workgroup reduce
    #pragma unroll
    for (int d = 16; d >= 1; d >>= 1) part += __shfl_down(part, d, 32);
    if (lane == 0) s_wave_loss[wave] = part;
    __syncthreads();
    if (tid == 0) {
        row_loss[row] = s_wave_loss[0] + s_wave_loss[1]
                      + s_wave_loss[2] + s_wave_loss[3];
    }
}

__global__ __launch_bounds__(256)
void ListMLE_mean_kernel(const float* __restrict__ row_loss,
                         float* __restrict__ out, int B)
{
    __shared__ float sm[256];
    float s = 0.0f;
    for (int i = threadIdx.x; i < B; i += 256) s += row_loss[i];
    sm[threadIdx.x] = s;
    __syncthreads();
    #pragma unroll
    for (int step = 128; step >= 1; step >>= 1) {
        if (threadIdx.x < step) sm[threadIdx.x] += sm[threadIdx.x + step];
        __syncthreads();
    }
    if (threadIdx.x == 0) out[0] = sm[0] * (1.0f / (float)B);
}

extern "C" void kernel_launch(void* const* d_in, const int* in_sizes, int n_in,
                              void* d_out, int out_size, void* d_ws, size_t ws_size,
                              hipStream_t stream) {
    const float* scores   = (const float*)d_in[0];
    const int*   rankings = (const int*)d_in[1];
    const int B = in_sizes[0] / N_ITEMS;

    float* row_loss = (float*)d_ws;       // B floats of scratch

    ListMLE_row_kernel<<<B, THREADS, 0, stream>>>(scores, rankings, row_loss);
    ListMLE_mean_kernel<<<1, 256, 0, stream>>>(row_loss, (float*)d_out, B);
}